// SpodNet_7095285973281
// MI455X (gfx1250) — compile-verified
//
#include <hip/hip_runtime.h>

#define PDIM 22
#define PM1  21
#define MPB  16        // matrices per block
#define TPB  256       // 8 waves of 32
#define MAT  484       // 22*22

typedef __attribute__((ext_vector_type(2))) float v2f;
typedef __attribute__((ext_vector_type(8))) float v8f;

__global__ __launch_bounds__(TPB, 1)
void spodnet_kernel(const float* __restrict__ ThetaIn,
                    const float* __restrict__ Wcol,
                    float* __restrict__ ThetaOut)
{
    __shared__ float Th[MPB * MAT];     // Theta, 16 matrices
    __shared__ float Wm[MPB * MAT];     // W = inv(Theta), updated in place
    __shared__ v2f   BmP[12 * 32];      // Wcol^T pair-interleaved: BmP[q*32+n] = {B[2q][n], B[2q+1][n]}
    __shared__ float t12pad[MPB * 24];  // gathered theta_12, zero-padded K->24
    __shared__ float t12n[MPB * 32];    // theta_12_next per matrix (stride 32, cols >=21 are 0)
    __shared__ float uu[MPB * 24];      // u = W11 @ t12n, later reused as w12_next
    __shared__ float w12s[MPB * 24];    // old w12 per matrix
    __shared__ float sc[MPB * 8];       // per-matrix scalars

    const int tid = threadIdx.x;
    const int m   = tid >> 4;           // matrix 0..15
    const int w   = tid & 15;           // worker within matrix
    const size_t gbase = (size_t)blockIdx.x * (MPB * MAT);

    float* Tb = &Th[m * MAT];
    float* Wb = &Wm[m * MAT];

    // ---- phase 0: load Theta (coalesced) into Th and Wm; build pair-interleaved Wcol^T ----
    for (int idx = tid; idx < MPB * MAT; idx += TPB) {
        float v = ThetaIn[gbase + idx];
        Th[idx] = v;
        Wm[idx] = v;
    }
    for (int idx = tid; idx < 12 * 32; idx += TPB) {
        int q = idx >> 5, n = idx & 31;       // q = K-pair index (K = 2q, 2q+1)
        int k0 = 2 * q, k1 = 2 * q + 1;
        v2f b;
        b.x = (k0 < PM1 && n < PM1) ? Wcol[n * PM1 + k0] : 0.0f;   // B[k][n] = Wcol[n][k]
        b.y = (k1 < PM1 && n < PM1) ? Wcol[n * PM1 + k1] : 0.0f;
        BmP[idx] = b;
    }
    __syncthreads();

    // ---- phase 1: Wm <- inv(Theta), in-place Gauss-Jordan (SPD, no pivoting) ----
    for (int k = 0; k < PDIM; ++k) {
        float p = 1.0f / Wb[k * PDIM + k];
        for (int j = w; j < PDIM; j += 16)
            if (j != k) Wb[k * PDIM + j] *= p;
        __syncthreads();
        for (int i = w; i < PDIM; i += 16) {
            if (i == k) continue;
            float f = Wb[i * PDIM + k];
            for (int j = 0; j < PDIM; ++j)
                if (j != k) Wb[i * PDIM + j] -= f * Wb[k * PDIM + j];
            Wb[i * PDIM + k] = -f * p;
        }
        if (w == 0) Wb[k * PDIM + k] = p;
        __syncthreads();
    }

    // ---- phase 2: one pass of per-column rank-1 updates ----
    for (int col = 0; col < PDIM; ++col) {
        // (a) gather old w12 and theta_12 (zero-padded), diag scalars
        for (int idx = w; idx < 24; idx += 16) {
            float vw = 0.0f, vt = 0.0f;
            if (idx < PM1) {
                int i = idx + (idx >= col);
                vw = Wb[i * PDIM + col];
                vt = Tb[i * PDIM + col];
            }
            w12s[m * 24 + idx]   = vw;
            t12pad[m * 24 + idx] = vt;
        }
        if (w == 0) {
            sc[m * 8 + 0] = Tb[col * PDIM + col];   // theta_22
            sc[m * 8 + 1] = Wb[col * PDIM + col];   // w_22
        }
        __syncthreads();

        // (b) wave 0: t12n(16x21) = theta12(16x24) @ Wcol^T(24x32) via WMMA f32 16x16x4
        if (tid < 32) {
            const int l  = tid;
            const int ml = l & 15;                  // M = matrix index (A) / N column (B)
            const int hi = l >> 4;                  // 0: K=k0,k0+1 ; 1: K=k0+2,k0+3
            const float* Arow = &t12pad[ml * 24 + 2 * hi];
            #pragma unroll
            for (int nt = 0; nt < 2; ++nt) {
                const int n = ml + nt * 16;
                v8f acc = {0.f, 0.f, 0.f, 0.f, 0.f, 0.f, 0.f, 0.f};
                #pragma unroll
                for (int kc = 0; kc < 6; ++kc) {
                    v2f a  = *(const v2f*)&Arow[kc * 4];        // K = kc*4 + 2*hi (+1)
                    v2f bb = BmP[(kc * 2 + hi) * 32 + n];
                    acc = __builtin_amdgcn_wmma_f32_16x16x4_f32(
                              false, a, false, bb, (short)0, acc, false, false);
                }
                #pragma unroll
                for (int r = 0; r < 8; ++r) {
                    int mm = r + hi * 8;
                    t12n[mm * 32 + n] = acc[r];     // cols >= 21 are exactly 0 (B padded)
                }
            }
        }
        __syncthreads();

        // (c) u = W11 @ t12n  (per matrix, old W values)
        for (int i = w; i < PM1; i += 16) {
            float s = 0.0f;
            int ir = (i + (i >= col)) * PDIM;
            for (int j = 0; j < PM1; ++j) {
                int jc = j + (j >= col);
                s += Wb[ir + jc] * t12n[m * 32 + j];
            }
            uu[m * 24 + i] = s;
        }
        __syncthreads();

        // (d) scalars: s1, schur, theta22_next, w22_next
        if (w == 0) {
            float w22 = sc[m * 8 + 1];
            float s1 = 0.0f;
            for (int j = 0; j < PM1; ++j) s1 += w12s[m * 24 + j] * t12n[m * 32 + j];
            float alpha = s1 / w22;
            float schur = 0.0f;
            for (int j = 0; j < PM1; ++j)
                schur += t12n[m * 32 + j] * (uu[m * 24 + j] - alpha * w12s[m * 24 + j]);
            float th22  = sc[m * 8 + 0];
            float th22n = (th22 - schur) + schur;   // matches reference expression
            float denom = th22n - schur;
            sc[m * 8 + 0] = th22n;
            sc[m * 8 + 1] = 1.0f / denom;           // w22_next
            sc[m * 8 + 2] = alpha;
            sc[m * 8 + 3] = 1.0f / w22;
            sc[m * 8 + 4] = denom;                  // 1 / w22_next
        }
        __syncthreads();

        // (e) Av -> w12_next (stored in uu), write Theta column/row
        {
            float w22n  = sc[m * 8 + 1];
            float alpha = sc[m * 8 + 2];
            for (int i = w; i < PM1; i += 16) {
                float av = uu[m * 24 + i] - alpha * w12s[m * 24 + i];
                uu[m * 24 + i] = -w22n * av;        // w12_next
                int ii = i + (i >= col);
                float tn = t12n[m * 32 + i];
                Tb[ii * PDIM + col] = tn;
                Tb[col * PDIM + ii] = tn;
            }
            if (w == 0) Tb[col * PDIM + col] = sc[m * 8 + 0];
        }
        __syncthreads();

        // (f) W update: W11 += w12n w12n^T / w22n - w12 w12^T / w22 ; set col/row/diag
        {
            float invw22 = sc[m * 8 + 3];
            float denom  = sc[m * 8 + 4];
            float w22n   = sc[m * 8 + 1];
            for (int idx = w; idx < PM1 * PM1; idx += 16) {
                int i = idx / PM1, j = idx - i * PM1;
                float wi = w12s[m * 24 + i], wj = w12s[m * 24 + j];
                float ni = uu[m * 24 + i],  nj = uu[m * 24 + j];
                int ii = (i + (i >= col)) * PDIM;
                int jj = j + (j >= col);
                Wb[ii + jj] += ni * nj * denom - wi * wj * invw22;
            }
            for (int i = w; i < PM1; i += 16) {
                int ii = i + (i >= col);
                float wn = uu[m * 24 + i];
                Wb[ii * PDIM + col] = wn;
                Wb[col * PDIM + ii] = wn;
            }
            if (w == 0) Wb[col * PDIM + col] = w22n;
        }
        __syncthreads();
    }

    // ---- phase 3: write Theta back (coalesced) ----
    for (int idx = tid; idx < MPB * MAT; idx += TPB) {
        ThetaOut[gbase + idx] = Th[idx];
    }
}

extern "C" void kernel_launch(void* const* d_in, const int* in_sizes, int n_in,
                              void* d_out, int out_size, void* d_ws, size_t ws_size,
                              hipStream_t stream) {
    const float* Theta = (const float*)d_in[0];
    const float* Wc    = (const float*)d_in[1];
    float* out = (float*)d_out;
    int Bn = in_sizes[0] / MAT;            // 16384
    int blocks = Bn / MPB;                 // 1024
    spodnet_kernel<<<dim3(blocks), dim3(TPB), 0, stream>>>(Theta, Wc, out);
}